// PhotoAndGeometryLoss_74191265071666
// MI455X (gfx1250) — compile-verified
//
#include <hip/hip_runtime.h>
#include <math.h>
#include <stdint.h>

#define B_    8
#define H_    256
#define W_    832
#define HW_   (H_*W_)
#define TW    38      // 32 + 2*3 halo
#define TH    14      // 8 + 2*3 halo
#define NPAIR 4
#define GX_   (W_/32) // 26
#define GY_   (H_/8)  // 32
#define NBLK  (GX_*GY_*B_)

#define C1_   (0.01f*0.01f)
#define C2_   (0.03f*0.03f)

__device__ __forceinline__ int reflect_idx(int i, int n) {
  i = (i < 0) ? -i : i;
  i = (i >= n) ? (2*n - 2 - i) : i;
  return i;
}

// CDNA5 async copy: global -> LDS, tracked by ASYNCcnt (no VGPR staging).
// LDS operand is the AS(3) byte offset = low 32 bits of the generic pointer.
__device__ __forceinline__ void async_ld_f32(const float* g, const float* lds_dst) {
  unsigned loff = (unsigned)(uintptr_t)lds_dst;
  asm volatile("global_load_async_to_lds_b32 %0, %1, off"
               :
               : "v"(loff), "v"(g)
               : "memory");
}

__device__ __forceinline__ void wait_async0() {
  asm volatile("s_wait_asynccnt 0x0" ::: "memory");
}

__device__ __forceinline__ void mat3mul(const float* A, const float* Bm, float* Cm) {
#pragma unroll
  for (int r = 0; r < 3; ++r)
#pragma unroll
    for (int c = 0; c < 3; ++c)
      Cm[r*3+c] = A[r*3+0]*Bm[0*3+c] + A[r*3+1]*Bm[1*3+c] + A[r*3+2]*Bm[2*3+c];
}

// ---------------------------------------------------------------------------
// Kernel 1: per (pair, batch) precompute A = K * R(euler) * K^-1 and b = K * t
// ---------------------------------------------------------------------------
__global__ void pose_prep_kernel(const float* __restrict__ k,
                                 const float* __restrict__ kinv,
                                 const float* __restrict__ poses,
                                 const float* __restrict__ poses_inv,
                                 float* __restrict__ wsA) {
  int t = threadIdx.x;
  if (t >= NPAIR * B_) return;
  int p = t >> 3, b = t & 7;
  int i = p >> 1;
  const float* pose = ((p & 1) ? poses_inv : poses) + (i*B_ + b)*6;
  float tx = pose[0], ty = pose[1], tz = pose[2];
  float ax = pose[3], ay = pose[4], az = pose[5];
  float cz = cosf(az), sz = sinf(az);
  float cy = cosf(ay), sy = sinf(ay);
  float cx = cosf(ax), sx = sinf(ax);
  float Zm[9] = {cz,-sz,0.f,  sz,cz,0.f,  0.f,0.f,1.f};
  float Ym[9] = {cy,0.f,sy,   0.f,1.f,0.f, -sy,0.f,cy};
  float Xm[9] = {1.f,0.f,0.f, 0.f,cx,-sx,  0.f,sx,cx};
  float YZ[9], R[9], RK[9], A[9];
  mat3mul(Ym, Zm, YZ);
  mat3mul(Xm, YZ, R);          // R = X*Y*Z
  const float* K  = k    + b*9;
  const float* Ki = kinv + b*9;
  mat3mul(R, Ki, RK);
  mat3mul(K, RK, A);           // A = K*R*K^-1
  float* o = wsA + (p*B_ + b)*12;
#pragma unroll
  for (int j = 0; j < 9; ++j) o[j] = A[j];
#pragma unroll
  for (int r = 0; r < 3; ++r)
    o[9+r] = K[r*3+0]*tx + K[r*3+1]*ty + K[r*3+2]*tz;   // b = K*t
}

// ---------------------------------------------------------------------------
// Kernel 2: fused warp + bilinear sample + separable SSIM(7x7 reflect) +
// masks + channel argmin + per-block partial reduction. Block = 32x8 tile.
// ---------------------------------------------------------------------------
__global__ void __launch_bounds__(256)
fused_loss_kernel(const float* __restrict__ tgt_img,
                  const float* __restrict__ ref_imgs,
                  const float* __restrict__ tgt_depth,
                  const float* __restrict__ ref_depths,
                  const float* __restrict__ wsA,
                  float* __restrict__ part) {
  __shared__ float sA[NPAIR][12];
  __shared__ float s_x[3][TH*TW];   // tgt channels over halo (async-loaded)
  __shared__ float s_y[3][TH*TW];   // warped-ref channels over halo
  __shared__ float s_pd[256];       // projected depth (center 8x32)
  __shared__ float s_cd[256];       // computed depth (center 8x32)
  __shared__ float s_col[5][8*TW];  // separable SSIM column sums
  __shared__ float s_red[3][256];

  const int tid = threadIdx.x;
  const int bx = blockIdx.x, by = blockIdx.y, b = blockIdx.z;
  const int w0 = bx*32, h0 = by*8;
  const int tx = tid & 31, ty = tid >> 5;
  const int hc = h0 + ty, wc = w0 + tx;

  if (tid < NPAIR*12) sA[tid/12][tid%12] = wsA[(tid/12)*B_*12 + b*12 + (tid%12)];

  float dcv[NPAIR], div_[NPAIR], ddv[NPAIR], vmv[NPAIR];

#pragma unroll 1
  for (int p = 0; p < NPAIR; ++p) {
    const float *tgtp, *dtp, *refp, *refdp;
    int RC;
    if (p == 0) { tgtp = tgt_img + b*3*HW_;           dtp = tgt_depth + b*HW_;
                  refp = ref_imgs + b*3*HW_;          refdp = ref_depths + b*HW_;          RC = 3; }
    else if (p == 1) { tgtp = ref_imgs + b*3*HW_;     dtp = ref_depths + b*HW_;
                  refp = tgt_depth + b*HW_;           refdp = tgt_depth + b*HW_;           RC = 1; }
    else if (p == 2) { tgtp = tgt_img + b*3*HW_;      dtp = tgt_depth + b*HW_;
                  refp = ref_imgs + (B_ + b)*3*HW_;   refdp = ref_depths + (B_ + b)*HW_;   RC = 3; }
    else        { tgtp = ref_imgs + (B_ + b)*3*HW_;   dtp = ref_depths + (B_ + b)*HW_;
                  refp = tgt_depth + b*HW_;           refdp = tgt_depth + b*HW_;           RC = 1; }

    __syncthreads();  // previous pair's compute done; sA visible on p==0

    __builtin_prefetch(tgtp + h0*W_ + w0, 0, 0);   // global_prefetch_b8
    __builtin_prefetch(refp + h0*W_ + w0, 0, 0);

    const float A0 = sA[p][0], A1 = sA[p][1], A2 = sA[p][2];
    const float A3 = sA[p][3], A4 = sA[p][4], A5 = sA[p][5];
    const float A6 = sA[p][6], A7 = sA[p][7], A8 = sA[p][8];
    const float b0 = sA[p][9], b1 = sA[p][10], b2 = sA[p][11];

    // ---- halo fill: async tgt-tile copy to LDS + warp/bilinear into LDS ----
    for (int idx = tid; idx < TH*TW; idx += 256) {
      int hy = idx / TW, hx = idx - hy*TW;
      int gh = reflect_idx(h0 - 3 + hy, H_);
      int gw = reflect_idx(w0 - 3 + hx, W_);
      int gidx = gh*W_ + gw;

      // async global->LDS staging of the three tgt channels (ASYNCcnt)
      async_ld_f32(tgtp + gidx,          &s_x[0][idx]);
      async_ld_f32(tgtp + HW_ + gidx,    &s_x[1][idx]);
      async_ld_f32(tgtp + 2*HW_ + gidx,  &s_x[2][idx]);

      float d  = dtp[gidx];
      float fx = (float)gw, fy = (float)gh;
      float pcx = (A0*fx + A1*fy + A2)*d + b0;
      float pcy = (A3*fx + A4*fy + A5)*d + b1;
      float pcz = (A6*fx + A7*fy + A8)*d + b2;
      float inv = 1.0f / (pcz + 1e-6f);
      float u = pcx*inv, v = pcy*inv;
      float gxn = (u / (float)(W_-1) - 0.5f) * 2.0f;
      float gyn = (v / (float)(H_-1) - 0.5f) * 2.0f;
      float xs  = ((gxn + 1.0f)*(float)W_ - 1.0f)*0.5f;
      float ysn = ((gyn + 1.0f)*(float)H_ - 1.0f)*0.5f;
      float x0f = floorf(xs), y0f = floorf(ysn);
      float wx = xs - x0f, wy = ysn - y0f;

      bool vx0 = (x0f >= 0.f) && (x0f < (float)W_);
      bool vx1 = (x0f+1.f >= 0.f) && (x0f+1.f < (float)W_);
      bool vy0 = (y0f >= 0.f) && (y0f < (float)H_);
      bool vy1 = (y0f+1.f >= 0.f) && (y0f+1.f < (float)H_);
      int ix0 = (int)fminf(fmaxf(x0f,      0.f), (float)(W_-1));
      int ix1 = (int)fminf(fmaxf(x0f+1.f,  0.f), (float)(W_-1));
      int iy0 = (int)fminf(fmaxf(y0f,      0.f), (float)(H_-1));
      int iy1 = (int)fminf(fmaxf(y0f+1.f,  0.f), (float)(H_-1));
      int o00 = iy0*W_+ix0, o10 = iy0*W_+ix1, o01 = iy1*W_+ix0, o11 = iy1*W_+ix1;
      float w00 = (1.f-wx)*(1.f-wy)*((vx0&&vy0)?1.f:0.f);
      float w10 =       wx*(1.f-wy)*((vx1&&vy0)?1.f:0.f);
      float w01 = (1.f-wx)*wy      *((vx0&&vy1)?1.f:0.f);
      float w11 =       wx*wy      *((vx1&&vy1)?1.f:0.f);

      float y0v = w00*refp[o00] + w10*refp[o10] + w01*refp[o01] + w11*refp[o11];
      float y1v, y2v;
      if (RC == 3) {
        y1v = w00*refp[HW_+o00]   + w10*refp[HW_+o10]   + w01*refp[HW_+o01]   + w11*refp[HW_+o11];
        y2v = w00*refp[2*HW_+o00] + w10*refp[2*HW_+o10] + w01*refp[2*HW_+o01] + w11*refp[2*HW_+o11];
      } else { y1v = y0v; y2v = y0v; }   // broadcast 1ch warped vs 3ch tgt
      s_y[0][idx] = y0v; s_y[1][idx] = y1v; s_y[2][idx] = y2v;

      if (hy >= 3 && hy < 11 && hx >= 3 && hx < 35) {
        float pdv = w00*refdp[o00] + w10*refdp[o10] + w01*refdp[o01] + w11*refdp[o11];
        int ci = (hy-3)*32 + (hx-3);
        s_pd[ci] = pdv;
        s_cd[ci] = pcz;
      }
    }
    wait_async0();    // drain this wave's ASYNCcnt before the barrier
    __syncthreads();

    // -------- per-center-pixel: separable SSIM + diffs + masks --------
    const int idx_c = (ty+3)*TW + (tx+3);
    float diffsum = 0.f, dcsum = 0.f, axsum = 0.f, aysum = 0.f, idsum = 0.f;
#pragma unroll 1
    for (int c = 0; c < 3; ++c) {
      __syncthreads();   // s_col reuse between channels
      // stage 1: 7-row column sums of {x, y, xx, yy, xy} over the halo
      for (int e = tid; e < 8*TW; e += 256) {
        int r = e / TW, x = e - r*TW;
        float csx = 0.f, csy = 0.f, csxx = 0.f, csyy = 0.f, csxy = 0.f;
#pragma unroll
        for (int dy = 0; dy < 7; ++dy) {
          float xv = s_x[c][(r+dy)*TW + x];
          float yv = s_y[c][(r+dy)*TW + x];
          csx += xv; csy += yv;
          csxx += xv*xv; csyy += yv*yv; csxy += xv*yv;
        }
        s_col[0][e] = csx; s_col[1][e] = csy;
        s_col[2][e] = csxx; s_col[3][e] = csyy; s_col[4][e] = csxy;
      }
      __syncthreads();
      // stage 2: 7-column sums -> full 7x7 window stats
      float sx_ = 0.f, sy_ = 0.f, sxx = 0.f, syy = 0.f, sxy = 0.f;
      int rb = ty*TW + tx;
#pragma unroll
      for (int dx = 0; dx < 7; ++dx) {
        sx_ += s_col[0][rb+dx]; sy_ += s_col[1][rb+dx];
        sxx += s_col[2][rb+dx]; syy += s_col[3][rb+dx]; sxy += s_col[4][rb+dx];
      }
      float mu_x = sx_ / 49.0f, mu_y = sy_ / 49.0f;
      float sig_x  = sxx/49.0f - mu_x*mu_x;
      float sig_y  = syy/49.0f - mu_y*mu_y;
      float sig_xy = sxy/49.0f - mu_x*mu_y;
      float nn = (2.f*mu_x*mu_y + C1_) * (2.f*sig_xy + C2_);
      float dd = (mu_x*mu_x + mu_y*mu_y + C1_) * (sig_x + sig_y + C2_);
      float ss = fminf(fmaxf((1.f - nn/dd)*0.5f, 0.f), 1.f);

      float xc = s_x[c][idx_c], yc = s_y[c][idx_c];
      float ad = fabsf(xc - yc);
      diffsum += 0.15f*fminf(ad, 1.0f) + 0.85f*ss;
      dcsum   += ad;
      axsum   += fabsf(xc);
      aysum   += fabsf(yc);
      float rc = refp[(RC == 3 ? c : 0)*HW_ + hc*W_ + wc];
      idsum   += fabsf(xc - rc);
    }

    float cdv = s_cd[tid], pdv = s_pd[tid];
    float ddep = fabsf(cdv - pdv) / (cdv + pdv + 1e-6f);
    float vcd  = (cdv > 0.01f && cdv < 100.0f) ? 1.f : 0.f;
    float vpd  = (pdv > 0.01f && pdv < 100.0f) ? 1.f : 0.f;
    float vref = ((aysum/3.0f) > 0.001f) ? 1.f : 0.f;
    float vtgt = ((axsum/3.0f) > 0.001f) ? 1.f : 0.f;
    float dcol = dcsum / 3.0f;
    float am   = (dcol < (idsum/3.0f)) ? 1.f : 0.f;

    dcv[p] = dcol;
    ddv[p] = ddep;
    vmv[p] = vtgt*vref*vpd*vcd*am;
    div_[p] = (diffsum/3.0f) * (1.0f - ddep);
  }

  // -------- argmin over pairs (first-min semantics), local contributions ----
  int j = 0;
#pragma unroll
  for (int p = 1; p < NPAIR; ++p) if (dcv[p] < dcv[j]) j = p;
  float lm = vmv[j];
  float lp = div_[j]*vmv[j];
  float lg = ddv[j]*vmv[j];

  // -------- deterministic block reduction --------
  __syncthreads();
  s_red[0][tid] = lm; s_red[1][tid] = lp; s_red[2][tid] = lg;
  __syncthreads();
  for (int s = 128; s > 0; s >>= 1) {
    if (tid < s) {
      s_red[0][tid] += s_red[0][tid+s];
      s_red[1][tid] += s_red[1][tid+s];
      s_red[2][tid] += s_red[2][tid+s];
    }
    __syncthreads();
  }
  if (tid == 0) {
    int blk = bx + GX_*(by + GY_*b);
    part[3*blk+0] = s_red[0][0];
    part[3*blk+1] = s_red[1][0];
    part[3*blk+2] = s_red[2][0];
  }
}

// ---------------------------------------------------------------------------
// Kernel 3: deterministic final reduction over block partials -> 2 scalars
// ---------------------------------------------------------------------------
__global__ void __launch_bounds__(256)
finalize_kernel(const float* __restrict__ part, float* __restrict__ out, int nblk) {
  __shared__ float r[3][256];
  int tid = threadIdx.x;
  float m = 0.f, ph = 0.f, g = 0.f;
  for (int i = tid; i < nblk; i += 256) {
    m  += part[3*i+0];
    ph += part[3*i+1];
    g  += part[3*i+2];
  }
  r[0][tid] = m; r[1][tid] = ph; r[2][tid] = g;
  __syncthreads();
  for (int s = 128; s > 0; s >>= 1) {
    if (tid < s) {
      r[0][tid] += r[0][tid+s];
      r[1][tid] += r[1][tid+s];
      r[2][tid] += r[2][tid+s];
    }
    __syncthreads();
  }
  if (tid == 0) {
    float ms = r[0][0];
    float dv = fmaxf(ms, 1.0f);
    out[0] = (ms > 100.f) ? (r[1][0] / dv) : 0.f;
    out[1] = (ms > 100.f) ? (r[2][0] / dv) : 0.f;
  }
}

extern "C" void kernel_launch(void* const* d_in, const int* in_sizes, int n_in,
                              void* d_out, int out_size, void* d_ws, size_t ws_size,
                              hipStream_t stream) {
  const float* tgt_img    = (const float*)d_in[0];
  const float* ref_imgs   = (const float*)d_in[1];
  const float* tgt_depth  = (const float*)d_in[2];
  const float* ref_depths = (const float*)d_in[3];
  const float* k          = (const float*)d_in[4];
  const float* k_inv      = (const float*)d_in[5];
  const float* poses      = (const float*)d_in[6];
  const float* poses_inv  = (const float*)d_in[7];

  float* ws   = (float*)d_ws;
  float* wsA  = ws;         // 4*8*12 = 384 floats
  float* part = ws + 512;   // 3*NBLK = 19968 floats  (total ws use ~80 KB)

  pose_prep_kernel<<<1, 32, 0, stream>>>(k, k_inv, poses, poses_inv, wsA);

  dim3 grid(GX_, GY_, B_);
  fused_loss_kernel<<<grid, 256, 0, stream>>>(tgt_img, ref_imgs, tgt_depth,
                                              ref_depths, wsA, part);

  finalize_kernel<<<1, 256, 0, stream>>>(part, (float*)d_out, NBLK);
}